// ComplexNetCustomParam_89739046683234
// MI455X (gfx1250) — compile-verified
//
#include <hip/hip_runtime.h>
#include <hip/hip_bf16.h>

// out[t,k] = sum_a x[t,a] * M_re[k,a].  Stage 1 builds a zero-padded
// Mpad[16][100] (rows 0,1 = M_re, rows 2..15 = 0) in d_ws so stage 2's
// WMMA B-operand loads are branch-free.  Stage 2 streams x through
// v_wmma_f32_16x16x4_f32: one wave32 per 16-row tile, 25 K-steps of 4.
// Roofline: 200 MB of x reads dominate -> ~8.8 us floor at 23.3 TB/s.

typedef __attribute__((ext_vector_type(2))) float v2f;
typedef __attribute__((ext_vector_type(8))) float v8f;

// ---- Stage 1: Mpad[n][a] = (n<2) ? sum_ij Pre*Ar - Pim*Ai : 0 ----
__global__ void cnet_compute_M(const float* __restrict__ A_real,
                               const float* __restrict__ A_imag,
                               const float* __restrict__ psi_real,
                               const float* __restrict__ psi_imag,
                               float* __restrict__ Mpad) {
    const int tid = threadIdx.x;

    float pr[10], pi[10];
#pragma unroll
    for (int i = 0; i < 10; ++i) {
        pr[i] = psi_real[i];
        pi[i] = psi_imag[i];
    }

    for (int idx = tid; idx < 16 * 100; idx += blockDim.x) {
        const int n = idx / 100;   // padded output column
        const int a = idx % 100;   // feature
        float acc = 0.0f;
        if (n < 2) {
            const float* Ar = A_real + (size_t)n * 10000 + a;  // ij stride 100
            const float* Ai = A_imag + (size_t)n * 10000 + a;
#pragma unroll 2
            for (int i = 0; i < 10; ++i) {
#pragma unroll
                for (int j = 0; j < 10; ++j) {
                    const float p_re = pr[i] * pr[j] + pi[i] * pi[j];
                    const float p_im = pr[i] * pi[j] - pi[i] * pr[j];
                    const int off = (i * 10 + j) * 100;
                    acc += p_re * Ar[off] - p_im * Ai[off];
                }
            }
        }
        Mpad[idx] = acc;
    }
}

// ---- Stage 2: out = x @ Mpad^T via WMMA f32 16x16x4 ----
// A 16x4 f32 layout: lane l holds x[row=l&15, k0 + 2*(l>>4) + {0,1}].
// B  4x16 f32 layout: lane l holds column n=l&15, K rows (l>>4)*2 and +1.
// C/D 16x16 f32: VGPR v -> row v (lanes 0..15) / row v+8 (lanes 16..31), N=lane&15.
__global__ void cnet_gemv_wmma(const float* __restrict__ x,
                               const float* __restrict__ Mpad,
                               float* __restrict__ out,
                               int batch, int ntiles) {
    const int lane = threadIdx.x & 31;
    const int tile = blockIdx.x * (blockDim.x >> 5) + (threadIdx.x >> 5);
    if (tile >= ntiles) return;            // uniform per-wave exit

    const int tileBase = tile * 16;
    const int n  = lane & 15;              // B/C/D column
    const int kk = (lane >> 4) * 2;        // K sub-offset within a 4-step

    // Branch-free B preload from the zero-padded matrix.
    const float* mp = Mpad + n * 100 + kk;
    v2f b[25];
#pragma unroll
    for (int s = 0; s < 25; ++s)
        b[s] = *(const v2f*)(mp + 4 * s);

    // Clamp row for loads so EXEC stays all-1s through the WMMAs.
    int row = tileBase + n;
    if (row > batch - 1) row = batch - 1;
    const float* xrow = x + (size_t)row * 100 + kk;

    v8f acc = {};
#pragma unroll
    for (int s = 0; s < 25; ++s) {
        v2f a = *(const v2f*)(xrow + 4 * s);   // aligned 8B load
        acc = __builtin_amdgcn_wmma_f32_16x16x4_f32(
            false, a, false, b[s], (short)0, acc, false, false);
    }

    // Pair columns 0 and 1 per row via lane swizzle (SWAPX1: xor=1,and=0x1f),
    // executed by the full wave, then lanes 0 and 16 emit 8 x b64 stores.
    float other[8];
#pragma unroll
    for (int v = 0; v < 8; ++v) {
        int o = __builtin_amdgcn_ds_swizzle(__builtin_bit_cast(int, acc[v]), 0x041F);
        other[v] = __builtin_bit_cast(float, o);
    }

    if (n == 0) {                           // lanes 0 (rows 0..7) and 16 (rows 8..15)
        const int rowBase = tileBase + (lane >> 4) * 8;
        float* orow = out + (size_t)rowBase * 2;
#pragma unroll
        for (int v = 0; v < 8; ++v) {
            if (rowBase + v < batch) {
                v2f d;
                d.x = acc[v];               // column 0 (this lane)
                d.y = other[v];             // column 1 (neighbor lane)
                *(v2f*)(orow + v * 2) = d;  // aligned 8B store of out[row][0:2]
            }
        }
    }
}

extern "C" void kernel_launch(void* const* d_in, const int* in_sizes, int n_in,
                              void* d_out, int out_size, void* d_ws, size_t ws_size,
                              hipStream_t stream) {
    const float* x        = (const float*)d_in[0];
    const float* A_real   = (const float*)d_in[1];
    const float* A_imag   = (const float*)d_in[2];
    const float* psi_real = (const float*)d_in[3];
    const float* psi_imag = (const float*)d_in[4];
    float* out  = (float*)d_out;
    float* Mpad = (float*)d_ws;   // 1600 floats (6.4 KB): [16][100], rows >=2 zero

    cnet_compute_M<<<1, 256, 0, stream>>>(A_real, A_imag, psi_real, psi_imag, Mpad);

    const int batch  = in_sizes[0] / 100;          // 500000
    const int ntiles = (batch + 15) / 16;          // 31250
    const int wavesPerBlock = 8;                   // 256 threads
    const int grid = (ntiles + wavesPerBlock - 1) / wavesPerBlock;
    cnet_gemv_wmma<<<grid, wavesPerBlock * 32, 0, stream>>>(x, Mpad, out, batch, ntiles);
}